// STGNNLayer_9517647528021
// MI455X (gfx1250) — compile-verified
//
#include <hip/hip_runtime.h>
#include <hip/hip_bf16.h>

// ---------------------------------------------------------------------------
// MTGNN STGNN layer for MI455X (gfx1250), fp32 WMMA path.
//
// Pipeline (all on `stream`):
//  1. conv_gate_kernel   : dilated-inception filter/gate conv, tanh*sigmoid
//                          -> buf0 laid out [B][V][C*506] (GEMM-friendly)
//  2. mlp_wmma_kernel p0 : Y(d_out) = W0 @ buf0 + bias        (WMMA f32)
//  3. diffusion_wmma     : buf1 = A^T @ buf0                  (WMMA f32)
//  4. mlp_wmma_kernel p1 : Y += W1 @ buf1
//  5. diffusion_wmma     : buf0 = A^T @ buf1
//  6. mlp_wmma_kernel p2 : Y += W2 @ buf0
//  7. stats_partial / stats_final : deterministic BN batch stats
//  8. bn_readout_kernel  : in-place BN on Y + relu-dot-mean -> X_skip
//
// Both WMMA kernels compute TWO 16-row output tiles per wave sharing one
// B-fragment, halving global B traffic (the dominant stream) per GEMM.
// ---------------------------------------------------------------------------

typedef __attribute__((ext_vector_type(2))) float v2f;
typedef __attribute__((ext_vector_type(8))) float v8f;

#define B_      16
#define CIN     32
#define V_      200
#define LIN     512
#define LOUT    506
#define C_      32
#define ROWLEN  (C_ * LOUT)     // 16192  floats per (b,v) row in diffusion layout
#define XBUF_B  (V_ * ROWLEN)   // 3238400 floats per batch (diffusion buffers)
#define YCH     (V_ * LOUT)     // 101200 floats per (b,channel) in output layout
#define YB      (C_ * YCH)      // 3238400 floats per batch (output layout)
#define NSTAT_BLK 64

struct ConvW {
    const float* fw[4]; const float* fb[4];
    const float* gw[4]; const float* gb[4];
};

// ---------------------------------------------------------------------------
// 1. Fused dilated-inception (kernels 2,3,6,7) filter+gate conv, tanh*sigmoid.
//    All branches read the same 7-tap window [l, l+6]; branch k uses the last
//    k taps (truncation offset 7-k).  Output layout: [B][V][c*506 + l].
// ---------------------------------------------------------------------------
__global__ __launch_bounds__(256) void conv_gate_kernel(
    const float* __restrict__ X, ConvW W, float* __restrict__ out)
{
    __shared__ float sX[CIN][72];                  // 70 columns used
    const int b = blockIdx.z, v = blockIdx.y, l0 = blockIdx.x * 64;
    const int tid = threadIdx.x;

    const float* xb = X + (size_t)b * CIN * V_ * LIN + (size_t)v * LIN;
    for (int idx = tid; idx < CIN * 70; idx += 256) {
        int ci = idx / 70, j = idx - ci * 70;
        int gl = l0 + j;
        sX[ci][j] = (gl < LIN) ? xb[(size_t)ci * V_ * LIN + gl] : 0.f;
    }
    __syncthreads();

    const int c = tid & 31, lrow = tid >> 5;       // 32 channels x 8 l-rows
    const int br = c >> 3, oc = c & 7;
    const int kk = (br == 0) ? 2 : (br == 1) ? 3 : (br == 2) ? 6 : 7;
    const int off = 7 - kk;
    const float* fw = W.fw[br] + (size_t)(oc * CIN) * kk;
    const float* gw = W.gw[br] + (size_t)(oc * CIN) * kk;

    float accF[8], accG[8];
    const float bf = W.fb[br][oc], bg = W.gb[br][oc];
#pragma unroll
    for (int j = 0; j < 8; ++j) { accF[j] = bf; accG[j] = bg; }

    for (int ci = 0; ci < CIN; ++ci) {
        const float* fwc = fw + ci * kk;
        const float* gwc = gw + ci * kk;
        for (int t = 0; t < kk; ++t) {
            const float wf = fwc[t], wg = gwc[t];
            const int base = off + t + lrow;
#pragma unroll
            for (int j = 0; j < 8; ++j) {
                float xv = sX[ci][base + 8 * j];
                accF[j] = fmaf(xv, wf, accF[j]);
                accG[j] = fmaf(xv, wg, accG[j]);
            }
        }
    }

    float* orow = out + (size_t)b * XBUF_B + (size_t)v * ROWLEN
                      + (size_t)c * LOUT + l0;
#pragma unroll
    for (int j = 0; j < 8; ++j) {
        int l = lrow + 8 * j;
        if (l0 + l < LOUT) {
            float f = tanhf(accF[j]);
            float g = 1.f / (1.f + __expf(-accG[j]));
            orow[l] = f * g;
        }
    }
}

// ---------------------------------------------------------------------------
// 2. Diffusion step via fp32 WMMA:  dst[b][w][n] = sum_k A[k][w] * src[b][k][n]
//    GEMM M=200 (pad 224 = 7 pairs of 16-row tiles), N=16192, K=200.
//    Each wave computes TWO 16x16 tiles (rows 32p..32p+31) sharing one
//    B-fragment per k-step; the 32-column A slice lives in LDS.
// ---------------------------------------------------------------------------
__global__ __launch_bounds__(128) void diffusion_wmma_kernel(
    const float* __restrict__ Asup, const float* __restrict__ src,
    float* __restrict__ dst)
{
    __shared__ float sA[200 * 32];            // sA[k*32+mm] = A[k][32p+mm]
    const int b = blockIdx.z, p = blockIdx.y; // pair of m-tiles
    const int w0 = p * 32;
    const int tid = threadIdx.x;

    for (int idx = tid; idx < 200 * 32; idx += 128) {
        int k = idx >> 5, mm = idx & 31;
        int w = w0 + mm; if (w > V_ - 1) w = V_ - 1;   // clamp; store guarded
        sA[idx] = Asup[k * V_ + w];
    }
    __syncthreads();

    const int lane = tid & 31, wave = tid >> 5;
    const int nt = blockIdx.x * 4 + wave;          // 253*4 = 1012 tiles exact
    const int n0 = nt * 16;
    const int lo = lane & 15, hi = lane >> 4;

    const float* colB = src + (size_t)b * XBUF_B + n0 + lo;
    v8f acc0 = {}, acc1 = {};

#pragma unroll 2
    for (int k0 = 0; k0 < 200; k0 += 4) {
        v2f a0, a1, bb;
        const int r0 = (k0 + 2 * hi) * 32 + lo;
        a0.x = sA[r0];
        a0.y = sA[r0 + 32];
        a1.x = sA[r0 + 16];
        a1.y = sA[r0 + 48];
        bb.x = colB[(size_t)(k0 + 2 * hi)     * ROWLEN];
        bb.y = colB[(size_t)(k0 + 2 * hi + 1) * ROWLEN];
        acc0 = __builtin_amdgcn_wmma_f32_16x16x4_f32(
            false, a0, false, bb, (short)0, acc0, false, false);
        acc1 = __builtin_amdgcn_wmma_f32_16x16x4_f32(
            false, a1, false, bb, (short)0, acc1, false, false);
    }

    float* outB = dst + (size_t)b * XBUF_B + n0 + lo;
#pragma unroll
    for (int r = 0; r < 8; ++r) {
        int w = w0 + 8 * hi + r;
        if (w < V_) outB[(size_t)w * ROWLEN] = acc0[r];
    }
#pragma unroll
    for (int r = 0; r < 8; ++r) {
        int w = w0 + 16 + 8 * hi + r;
        if (w < V_) outB[(size_t)w * ROWLEN] = acc1[r];
    }
}

// ---------------------------------------------------------------------------
// 3. MLP (1x1 conv) accumulate pass via fp32 WMMA:
//    Y[b][o][v][l] (+)= sum_c mlp_w[o][wOff+c] * src[b][v][c][l]
//    M=32 (both tiles in one wave), K=32 (8 k-steps), N = V*LOUT = 101200.
// ---------------------------------------------------------------------------
__global__ __launch_bounds__(128) void mlp_wmma_kernel(
    const float* __restrict__ mlpW, const float* __restrict__ mlpB,
    const float* __restrict__ src, float* __restrict__ Y,
    int wOff, int accumulate)
{
    __shared__ float sW[32 * 32];             // sW[k*32+mm] = W[mm][wOff+k]
    const int b = blockIdx.z;
    const int tid = threadIdx.x;

    for (int idx = tid; idx < 32 * 32; idx += 128) {
        int k = idx >> 5, mm = idx & 31;
        sW[idx] = mlpW[mm * 96 + wOff + k];
    }
    __syncthreads();

    const int lane = tid & 31, wave = tid >> 5;
    const int nt = blockIdx.x * 4 + wave;
    if (nt >= 6325) return;                   // whole-wave exit, EXEC safe
    const int n0 = nt * 16;
    const int lo = lane & 15, hi = lane >> 4;

    const int n = n0 + lo;
    const int v = n / LOUT;
    const int l = n - v * LOUT;
    const float* colB = src + (size_t)b * XBUF_B + (size_t)v * ROWLEN + l;

    float* yB = Y + (size_t)b * YB + n0 + lo;
    v8f acc0, acc1;
    if (accumulate) {
#pragma unroll
        for (int r = 0; r < 8; ++r) {
            acc0[r] = yB[(size_t)(8 * hi + r) * YCH];
            acc1[r] = yB[(size_t)(16 + 8 * hi + r) * YCH];
        }
    } else {
#pragma unroll
        for (int r = 0; r < 8; ++r) {
            acc0[r] = mlpB[8 * hi + r];
            acc1[r] = mlpB[16 + 8 * hi + r];
        }
    }

#pragma unroll
    for (int k0 = 0; k0 < 32; k0 += 4) {
        v2f a0, a1, bb;
        const int r0 = (k0 + 2 * hi) * 32 + lo;
        a0.x = sW[r0];
        a0.y = sW[r0 + 32];
        a1.x = sW[r0 + 16];
        a1.y = sW[r0 + 48];
        bb.x = colB[(size_t)(k0 + 2 * hi)     * LOUT];
        bb.y = colB[(size_t)(k0 + 2 * hi + 1) * LOUT];
        acc0 = __builtin_amdgcn_wmma_f32_16x16x4_f32(
            false, a0, false, bb, (short)0, acc0, false, false);
        acc1 = __builtin_amdgcn_wmma_f32_16x16x4_f32(
            false, a1, false, bb, (short)0, acc1, false, false);
    }

#pragma unroll
    for (int r = 0; r < 8; ++r) {
        yB[(size_t)(8 * hi + r) * YCH]      = acc0[r];
        yB[(size_t)(16 + 8 * hi + r) * YCH] = acc1[r];
    }
}

// ---------------------------------------------------------------------------
// 4. BN batch stats: deterministic two-stage reduction (no float atomics).
// ---------------------------------------------------------------------------
__global__ __launch_bounds__(256) void stats_partial_kernel(
    const float* __restrict__ Y, float* __restrict__ psum, float* __restrict__ psq)
{
    const int c = blockIdx.y, blk = blockIdx.x, tid = threadIdx.x;
    float s = 0.f, q = 0.f;
    for (int b = 0; b < B_; ++b) {
        const float* p = Y + (size_t)b * YB + (size_t)c * YCH;
        for (int i = blk * 256 + tid; i < YCH; i += NSTAT_BLK * 256) {
            float y = p[i];
            s += y; q += y * y;
        }
    }
    __shared__ float ls[256], lq[256];
    ls[tid] = s; lq[tid] = q; __syncthreads();
    for (int st = 128; st > 0; st >>= 1) {
        if (tid < st) { ls[tid] += ls[tid + st]; lq[tid] += lq[tid + st]; }
        __syncthreads();
    }
    if (tid == 0) {
        psum[c * NSTAT_BLK + blk] = ls[0];
        psq [c * NSTAT_BLK + blk] = lq[0];
    }
}

__global__ void stats_final_kernel(
    const float* __restrict__ psum, const float* __restrict__ psq,
    const float* __restrict__ gamma, const float* __restrict__ beta,
    float* __restrict__ scale, float* __restrict__ shift)
{
    int c = threadIdx.x;
    if (c < C_) {
        float s = 0.f, q = 0.f;
        for (int i = 0; i < NSTAT_BLK; ++i) {
            s += psum[c * NSTAT_BLK + i];
            q += psq [c * NSTAT_BLK + i];
        }
        const float N = (float)((size_t)B_ * YCH);
        float mean = s / N;
        float var  = q / N - mean * mean;
        float rs   = rsqrtf(var + 1e-5f);
        float sc   = gamma[c] * rs;
        scale[c] = sc;
        shift[c] = beta[c] - mean * sc;
    }
}

// ---------------------------------------------------------------------------
// 5. In-place BN + readout: xbn = y*scale+shift (written back to d_out),
//    skip[b,v] = mean_l relu( sum_c end_w[c]*xbn + end_b ).
// ---------------------------------------------------------------------------
__global__ __launch_bounds__(256) void bn_readout_kernel(
    float* __restrict__ Y, const float* __restrict__ scale,
    const float* __restrict__ shift, const float* __restrict__ endW,
    const float* __restrict__ endB, float* __restrict__ skip)
{
    const int v = blockIdx.x, b = blockIdx.y, tid = threadIdx.x;
    __shared__ float sSc[32], sSh[32], sEw[32];
    if (tid < 32) { sSc[tid] = scale[tid]; sSh[tid] = shift[tid]; sEw[tid] = endW[tid]; }
    __syncthreads();

    const float eb = endB[0];
    float local = 0.f;
    float* base = Y + (size_t)b * YB + (size_t)v * LOUT;
    for (int l = tid; l < LOUT; l += 256) {
        float dot = eb;
#pragma unroll
        for (int c = 0; c < C_; ++c) {
            float y  = base[(size_t)c * YCH + l];
            float xb = fmaf(y, sSc[c], sSh[c]);
            base[(size_t)c * YCH + l] = xb;
            dot = fmaf(sEw[c], xb, dot);
        }
        local += fmaxf(dot, 0.f);
    }
    __shared__ float red[256];
    red[tid] = local; __syncthreads();
    for (int st = 128; st > 0; st >>= 1) {
        if (tid < st) red[tid] += red[tid + st];
        __syncthreads();
    }
    if (tid == 0) skip[b * V_ + v] = red[0] * (1.f / (float)LOUT);
}

// ---------------------------------------------------------------------------
extern "C" void kernel_launch(void* const* d_in, const int* in_sizes, int n_in,
                              void* d_out, int out_size, void* d_ws, size_t ws_size,
                              hipStream_t stream)
{
    (void)in_sizes; (void)n_in; (void)out_size; (void)ws_size;

    const float* X = (const float*)d_in[0];
    const float* A = (const float*)d_in[1];
    ConvW W;
    for (int i = 0; i < 4; ++i) {
        W.fw[i] = (const float*)d_in[2 + i];
        W.fb[i] = (const float*)d_in[6 + i];
        W.gw[i] = (const float*)d_in[10 + i];
        W.gb[i] = (const float*)d_in[14 + i];
    }
    const float* mlpW  = (const float*)d_in[18];
    const float* mlpB  = (const float*)d_in[19];
    const float* gamma = (const float*)d_in[20];
    const float* beta  = (const float*)d_in[21];
    const float* endW  = (const float*)d_in[22];
    const float* endB  = (const float*)d_in[23];

    float* Yout = (float*)d_out;                      // [B,32,V,506]
    float* skip = Yout + (size_t)B_ * YB;             // [B,V]

    float* buf0  = (float*)d_ws;
    float* buf1  = buf0 + (size_t)B_ * XBUF_B;
    float* psum  = buf1 + (size_t)B_ * XBUF_B;
    float* psq   = psum + C_ * NSTAT_BLK;
    float* scale = psq  + C_ * NSTAT_BLK;
    float* shift = scale + C_;

    // 1. gated temporal conv -> buf0 [B][V][C*506]
    conv_gate_kernel<<<dim3(8, V_, B_), 256, 0, stream>>>(X, W, buf0);
    // 2. Y = W0 @ buf0 + bias
    mlp_wmma_kernel<<<dim3(1582, 1, B_), 128, 0, stream>>>(mlpW, mlpB, buf0, Yout, 0, 0);
    // 3. buf1 = A^T @ buf0
    diffusion_wmma_kernel<<<dim3(253, 7, B_), 128, 0, stream>>>(A, buf0, buf1);
    // 4. Y += W1 @ buf1
    mlp_wmma_kernel<<<dim3(1582, 1, B_), 128, 0, stream>>>(mlpW, mlpB, buf1, Yout, 32, 1);
    // 5. buf0 = A^T @ buf1
    diffusion_wmma_kernel<<<dim3(253, 7, B_), 128, 0, stream>>>(A, buf1, buf0);
    // 6. Y += W2 @ buf0
    mlp_wmma_kernel<<<dim3(1582, 1, B_), 128, 0, stream>>>(mlpW, mlpB, buf0, Yout, 64, 1);
    // 7-8. deterministic BN stats
    stats_partial_kernel<<<dim3(NSTAT_BLK, C_), 256, 0, stream>>>(Yout, psum, psq);
    stats_final_kernel<<<1, 32, 0, stream>>>(psum, psq, gamma, beta, scale, shift);
    // 9. in-place BN + relu-mean readout
    bn_readout_kernel<<<dim3(V_, B_), 256, 0, stream>>>(Yout, scale, shift, endW, endB, skip);
}